// QueryAndGroup_11450382811203
// MI455X (gfx1250) — compile-verified
//
#include <hip/hip_runtime.h>

// Problem constants (fixed by the reference harness)
#define BQ_B 4
#define BQ_N 8192
#define BQ_P 2048
#define BQ_C 64
#define BQ_S 32
#define BQ_R2 (0.12f * 0.12f)

typedef __attribute__((ext_vector_type(2))) float v2f;
typedef __attribute__((ext_vector_type(8))) float v8f;

// ---------------------------------------------------------------------------
// Pass 0: stage db points as B-matrix columns: (-2x, -2y, -2z, ||x||^2)
// ---------------------------------------------------------------------------
__global__ void __launch_bounds__(256)
stage_points_kernel(const float* __restrict__ xyz, float4* __restrict__ stage) {
    int i = blockIdx.x * blockDim.x + threadIdx.x;
    if (i >= BQ_B * BQ_N) return;
    float x = xyz[3 * i + 0];
    float y = xyz[3 * i + 1];
    float z = xyz[3 * i + 2];
    stage[i] = make_float4(-2.0f * x, -2.0f * y, -2.0f * z, x * x + y * y + z * z);
}

// ---------------------------------------------------------------------------
// Pass 1: ball query. One wave32 handles 16 query points. d2 tiles (16 queries
// x 16 db points each) computed by V_WMMA_F32_16X16X4_F32:
//   A row m  = (qx, qy, qz, 1)               [16x4]
//   B col n  = (-2x, -2y, -2z, ||x||^2)      [4x16]  (staged)
//   C[m][n]  = ||q_m||^2
//   D = A*B + C = ||q-x||^2  exactly.
// 32 db columns per iteration (2 WMMAs), software-pipelined B-fragment loads,
// near-cache prefetch one iteration ahead.
// ---------------------------------------------------------------------------
__global__ void __launch_bounds__(256)
ball_query_wmma_kernel(const float* __restrict__ new_xyz,
                       const float4* __restrict__ stage,
                       int* __restrict__ idx_out) {
    const int lane  = threadIdx.x & 31;
    const int wave  = blockIdx.x * (blockDim.x >> 5) + (threadIdx.x >> 5);
    const int qbase = wave * 16;              // 16 queries per wave; P%16==0 so
    const int b     = qbase / BQ_P;           // all 16 queries share one batch
    const int m     = lane & 15;              // query row owned by this lane

    // Load this row's query point and ||q||^2
    const float* q = new_xyz + 3 * (qbase + m);
    const float qx = q[0], qy = q[1], qz = q[2];
    const float qq = qx * qx + qy * qy + qz * qz;

    // A-matrix fragment (16x4 f32, 2 VGPRs)
    v2f a;
    if (lane < 16) { a.x = qx; a.y = qy; }
    else           { a.x = qz; a.y = 1.0f; }

    // C fragment: C[m][n] = ||q_m||^2 ; VGPR r holds row r (+8 for hi lanes)
    v8f c;
    const int rowOfs = (lane < 16) ? 0 : 8;
#pragma unroll
    for (int r = 0; r < 8; ++r) c[r] = __shfl(qq, r + rowOfs, 32);

    // Per-lane base of B-fragment data: column (m) component pair (kOfs)
    const float* bs  = (const float*)(stage + (size_t)b * BQ_N);
    const float* pc0 = bs + 4 * m + ((lane < 16) ? 0 : 2);

    int cnt = 0, first = 0;
    int* my_out = idx_out + (size_t)(qbase + m) * BQ_S;

    // Software pipeline: fragments for columns [0,16) and [16,32)
    v2f b0 = *(const v2f*)(pc0);
    v2f b1 = *(const v2f*)(pc0 + 64);

    for (int nb = 0; nb < BQ_N; nb += 32) {
        // Issue next iteration's fragment loads before consuming this one.
        // Clamp (scalar select) so the last iteration stays in the staged region.
        int nnext = (nb + 32 < BQ_N) ? (nb + 32) : 0;
        v2f n0 = *(const v2f*)(pc0 + 4 * nnext);
        v2f n1 = *(const v2f*)(pc0 + 4 * nnext + 64);
        __builtin_prefetch(pc0 + 4 * nnext + 128, 0, 3);  // near-cache prefetch

        // 512 pair distances in two instructions
        v8f d0 = __builtin_amdgcn_wmma_f32_16x16x4_f32(
            false, a, false, b0, (short)0, c, false, false);
        v8f d1 = __builtin_amdgcn_wmma_f32_16x16x4_f32(
            false, a, false, b1, (short)0, c, false, false);

        // Per-row 32-bit in-ball mask for columns nb..nb+31, ascending order.
        // ballot(d[r] < r2): bits 0-15 -> row r, bits 16-31 -> row r+8.
        unsigned int lo16 = 0u, hi16 = 0u;
#pragma unroll
        for (int r = 0; r < 8; ++r) {
            unsigned int bal0 = (unsigned int)__ballot(d0[r] < BQ_R2);
            unsigned int bal1 = (unsigned int)__ballot(d1[r] < BQ_R2);
            lo16 = (m == r) ? (bal0 & 0xFFFFu) : ((m == r + 8) ? (bal0 >> 16) : lo16);
            hi16 = (m == r) ? (bal1 & 0xFFFFu) : ((m == r + 8) ? (bal1 >> 16) : hi16);
        }
        unsigned int mask = lo16 | (hi16 << 16);

        // Append first-in-index-order hits (lanes 16-31 mirror lanes 0-15 so
        // cnt stays wave-consistent; only the low half stores).
        while (mask != 0u && cnt < BQ_S) {
            int bit = __ffs(mask) - 1;
            int n = nb + bit;
            if (cnt == 0) first = n;
            if (lane < 16) my_out[cnt] = n;
            ++cnt;
            mask &= mask - 1u;
        }

        if (__all(cnt >= BQ_S)) break;   // uniform early exit

        b0 = n0;
        b1 = n1;
    }

    // Tail fill: repeat first found index (0 if none found -> first==0, cnt==0)
    for (int s = cnt; s < BQ_S; ++s)
        if (lane < 16) my_out[s] = first;
}

// ---------------------------------------------------------------------------
// Pass 2: gather/group. One block per (b,p). Output [B, 3+C, P, S]:
//   c' < 3 : xyz[b, idx, c'] - new_xyz[b, p, c']
//   c' >= 3: features[b, c'-3, idx]
// Writes are fully coalesced over (c', s); gathered reads hit L2-resident rows.
// ---------------------------------------------------------------------------
__global__ void __launch_bounds__(256)
group_kernel(const float* __restrict__ xyz,
             const float* __restrict__ new_xyz,
             const float* __restrict__ feat,
             const int* __restrict__ idx,
             float* __restrict__ out) {
    const int bp = blockIdx.x;            // 0 .. B*P-1
    const int b  = bp / BQ_P;
    const int p  = bp % BQ_P;

    __shared__ int   sidx[BQ_S];
    __shared__ float sq[3];
    if (threadIdx.x < BQ_S) sidx[threadIdx.x] = idx[(size_t)bp * BQ_S + threadIdx.x];
    if (threadIdx.x < 3)    sq[threadIdx.x]   = new_xyz[(size_t)bp * 3 + threadIdx.x];
    __syncthreads();

    const int CT = 3 + BQ_C;              // 67 output channels
    for (int t = threadIdx.x; t < CT * BQ_S; t += blockDim.x) {
        int s  = t & (BQ_S - 1);
        int cc = t >> 5;
        int n  = sidx[s];
        float v;
        if (cc < 3) v = xyz[((size_t)b * BQ_N + n) * 3 + cc] - sq[cc];
        else        v = feat[(size_t)(b * BQ_C + (cc - 3)) * BQ_N + n];
        out[(((size_t)b * CT + cc) * BQ_P + p) * BQ_S + s] = v;
    }
}

// ---------------------------------------------------------------------------
extern "C" void kernel_launch(void* const* d_in, const int* in_sizes, int n_in,
                              void* d_out, int out_size, void* d_ws, size_t ws_size,
                              hipStream_t stream) {
    const float* xyz     = (const float*)d_in[0];   // [B,N,3]
    const float* new_xyz = (const float*)d_in[1];   // [B,P,3]
    const float* feat    = (const float*)d_in[2];   // [B,C,N]
    float*       out     = (float*)d_out;           // [B,3+C,P,S]

    // Workspace layout: [stage float4 B*N][idx int32 B*P*S]
    float4* stage = (float4*)d_ws;
    int*    idx   = (int*)((char*)d_ws + (size_t)BQ_B * BQ_N * sizeof(float4));

    // Pass 0: stage B-matrix columns
    {
        int total = BQ_B * BQ_N;
        stage_points_kernel<<<(total + 255) / 256, 256, 0, stream>>>(xyz, stage);
    }

    // Pass 1: WMMA ball query. 16 queries per wave -> B*P/16 = 512 waves.
    {
        int waves  = (BQ_B * BQ_P) / 16;   // 512
        int wavesPerBlock = 8;             // 256 threads
        ball_query_wmma_kernel<<<waves / wavesPerBlock, wavesPerBlock * 32, 0, stream>>>(
            new_xyz, stage, idx);
    }

    // Pass 2: gather/group (HBM-bound phase)
    group_kernel<<<BQ_B * BQ_P, 256, 0, stream>>>(xyz, new_xyz, feat, idx, out);
}